// AE_gnnrnn_gpu_57002805953278
// MI455X (gfx1250) — compile-verified
//
#include <hip/hip_runtime.h>
#include <hip/hip_bf16.h>
#include <stdint.h>

typedef float    v2f __attribute__((ext_vector_type(2)));
typedef float    v8f __attribute__((ext_vector_type(8)));
typedef _Float16 h4v __attribute__((ext_vector_type(4)));

static constexpr int NROWS = 4096;
static constexpr int SEQL  = 64;
static constexpr int HID   = 13;
static constexpr int G4    = 52;   // 4*H
static constexpr int DD2   = 26;   // 2*H
static constexpr int INDIM = 64;
static constexpr int VOCAB = 32000;

// ---------------- workspace layout (bytes, all 256-aligned) ----------------
static constexpr size_t OFF_XG   = 0;                       // 4096*64*52 f16  = 27,262,976
static constexpr size_t OFF_LAST = 27262976;                // 4096*26 f32     = 425,984
static constexpr size_t OFF_SH   = 27688960;                // 425,984
static constexpr size_t OFF_SC   = 28114944;                // 425,984
static constexpr size_t OFF_DEG  = 28540928;                // 16,384
static constexpr size_t OFF_NORM = 28557312;                // 16,384
static constexpr size_t OFF_XW   = 28573696;                // 4096*32 f32     = 524,288
static constexpr size_t OFF_AGG  = 29097984;                // 524,288
static constexpr size_t OFF_HG   = 29622272;                // 425,984
static constexpr size_t OFF_CG   = 30048256;                // 425,984
static constexpr size_t OFF_DEC  = 30474240;                // 425,984

__device__ __forceinline__ float sigf(float v)     { return 1.0f / (1.0f + __expf(-v)); }
__device__ __forceinline__ float tanhfast(float v) { return 2.0f / (1.0f + __expf(-2.0f * v)) - 1.0f; }
__device__ __forceinline__ float leakyf(float v)   { return v > 0.0f ? v : 0.01f * v; }

// ============================================================================
// K1: xg = gather(emb, x) @ w_ih^T + b  — f32 WMMA 16x16x4, one M-tile/wave.
// A: 16 gathered emb rows (K=64, 16 k-steps); B: w_ih (52 cols padded to 64).
// Output stored compactly as f16 at stride 52.
// ============================================================================
__global__ void k_xg_wmma(const int* __restrict__ x, const float* __restrict__ emb,
                          const float* __restrict__ w_ih, const float* __restrict__ bias,
                          _Float16* __restrict__ xg)
{
    const int gid  = blockIdx.x * blockDim.x + threadIdx.x;
    const int wave = gid >> 5;                 // M-tile index (16 rows of N*L)
    const int lane = threadIdx.x & 31;
    const int col  = lane & 15;
    const int koff = (lane & 16) >> 3;         // hi half-wave handles K+2,K+3

    const int row = wave * 16 + col;           // flat (n*64 + l)
    const float* arow = emb + (size_t)x[row] * INDIM;

    v2f av[16];
#pragma unroll
    for (int s = 0; s < 16; ++s) {
        const int k = s * 4 + koff;
        av[s].x = arow[k];
        av[s].y = arow[k + 1];
    }

    const int mbase = wave * 16 + ((lane & 16) >> 1);   // +8 rows for hi half
#pragma unroll
    for (int t = 0; t < 4; ++t) {
        const int  g  = t * 16 + col;
        const bool gv = (g < G4);
        const float* wrow = w_ih + (size_t)(gv ? g : 0) * INDIM;
        v8f acc = {0.f, 0.f, 0.f, 0.f, 0.f, 0.f, 0.f, 0.f};
#pragma unroll
        for (int s = 0; s < 16; ++s) {
            const int k = s * 4 + koff;
            const float b0 = wrow[k];
            const float b1 = wrow[k + 1];
            v2f bv; bv.x = gv ? b0 : 0.f; bv.y = gv ? b1 : 0.f;
            acc = __builtin_amdgcn_wmma_f32_16x16x4_f32(
                false, av[s], false, bv, (short)0, acc, false, false);
        }
        if (gv) {
            const float bg = bias[g];
#pragma unroll
            for (int v = 0; v < 8; ++v)
                xg[(size_t)(mbase + v) * G4 + g] = (_Float16)(acc[v] + bg);
        }
    }
}

// ============================================================================
// K2: forward LSTM scan, one lane per sequence row. Keeps h,c in registers,
// w_hh is wave-uniform -> scalar loads. Writes final h to last[:, 0:13].
// ============================================================================
__global__ void k_scan_fwd(const _Float16* __restrict__ xg, const float* __restrict__ w_hh,
                           float* __restrict__ last)
{
    const int row = blockIdx.x * blockDim.x + threadIdx.x;
    if (row >= NROWS) return;
    float h[HID], c[HID];
#pragma unroll
    for (int j = 0; j < HID; ++j) { h[j] = 0.f; c[j] = 0.f; }

    for (int t = 0; t < SEQL; ++t) {
        const h4v* p = (const h4v*)(xg + ((size_t)row * SEQL + t) * G4);
        float gates[G4];
#pragma unroll
        for (int q = 0; q < 13; ++q) {
            h4v v4 = p[q];
            gates[4*q + 0] = (float)v4.x; gates[4*q + 1] = (float)v4.y;
            gates[4*q + 2] = (float)v4.z; gates[4*q + 3] = (float)v4.w;
        }
#pragma unroll
        for (int g = 0; g < G4; ++g) {
            float acc = gates[g];
#pragma unroll
            for (int k = 0; k < HID; ++k) acc += h[k] * w_hh[g * HID + k];
            gates[g] = acc;
        }
#pragma unroll
        for (int j = 0; j < HID; ++j) {
            const float iv = sigf(gates[j]);
            const float fv = sigf(gates[HID + j]);
            const float gv = tanhfast(gates[2 * HID + j]);
            const float ov = sigf(gates[3 * HID + j]);
            c[j] = fv * c[j] + iv * gv;
            h[j] = tanhfast(c[j]) * ov;
        }
    }
#pragma unroll
    for (int j = 0; j < HID; ++j) last[(size_t)row * DD2 + j] = h[j];
}

// ============================================================================
// K3: backward LSTM = single cell step on embs[:,63] (hb[:,-1] == y_b[:,0]),
// then both 26x26 projections (sh, sc).
// ============================================================================
__global__ void k_bwd_proj(const int* __restrict__ x, const float* __restrict__ emb,
                           const float* __restrict__ w_ih_b, const float* __restrict__ b_b,
                           const float* __restrict__ p1w, const float* __restrict__ p1b,
                           const float* __restrict__ p2w, const float* __restrict__ p2b,
                           const float* __restrict__ last_f,
                           float* __restrict__ sh, float* __restrict__ sc)
{
    const int n = blockIdx.x * blockDim.x + threadIdx.x;
    if (n >= NROWS) return;
    const float* e = emb + (size_t)x[n * SEQL + (SEQL - 1)] * INDIM;

    float lastv[DD2];
#pragma unroll
    for (int j = 0; j < HID; ++j) lastv[j] = last_f[(size_t)n * DD2 + j];

    float gates[G4];
    for (int g = 0; g < G4; ++g) {
        float acc = b_b[g];
        for (int k = 0; k < INDIM; ++k) acc += e[k] * w_ih_b[g * INDIM + k];
        gates[g] = acc;
    }
#pragma unroll
    for (int j = 0; j < HID; ++j) {
        const float iv = sigf(gates[j]);
        const float gv = tanhfast(gates[2 * HID + j]);
        const float ov = sigf(gates[3 * HID + j]);
        const float cc = iv * gv;          // c0 = 0, forget term vanishes
        lastv[HID + j] = tanhfast(cc) * ov;
    }
    for (int o = 0; o < DD2; ++o) {
        float s1 = p1b[o], s2 = p2b[o];
        for (int k = 0; k < DD2; ++k) {
            s1 += lastv[k] * p1w[o * DD2 + k];
            s2 += lastv[k] * p2w[o * DD2 + k];
        }
        sh[(size_t)n * DD2 + o] = s1;
        sc[(size_t)n * DD2 + o] = s2;
    }
}

// ============================================================================
// Graph degree / norm
// ============================================================================
__global__ void k_deg_init(float* __restrict__ deg)
{
    const int n = blockIdx.x * blockDim.x + threadIdx.x;
    if (n < NROWS) deg[n] = 1.0f;
}
__global__ void k_deg_scatter(const int* __restrict__ ei, float* __restrict__ deg, int E)
{
    const int e = blockIdx.x * blockDim.x + threadIdx.x;
    if (e < E) atomicAdd(&deg[ei[E + e]], 1.0f);
}
__global__ void k_norm(const float* __restrict__ deg, float* __restrict__ norm)
{
    const int n = blockIdx.x * blockDim.x + threadIdx.x;
    if (n < NROWS) norm[n] = rsqrtf(deg[n]);
}

// ============================================================================
// GNN helpers: per-row matmul (templated, register-resident), agg init, scatter
// ============================================================================
template <int DIN, int DOUT, bool LEAKY>
__global__ void k_rowmm(const float* __restrict__ xin, const float* __restrict__ w,
                        const float* __restrict__ bias, float* __restrict__ y)
{
    const int n = blockIdx.x * blockDim.x + threadIdx.x;
    if (n >= NROWS) return;
    float xv[DIN];
#pragma unroll
    for (int k = 0; k < DIN; ++k) {
        float v = xin[(size_t)n * DIN + k];
        if (LEAKY) v = leakyf(v);
        xv[k] = v;
    }
#pragma unroll
    for (int o = 0; o < DOUT; ++o) {
        float acc = bias ? bias[o] : 0.0f;
#pragma unroll
        for (int k = 0; k < DIN; ++k) acc += xv[k] * w[o * DIN + k];
        y[(size_t)n * DOUT + o] = acc;
    }
}

__global__ void k_agg_init(const float* __restrict__ xw, const float* __restrict__ norm,
                           const float* __restrict__ b, float* __restrict__ agg,
                           int dlog, int total)
{
    const int i = blockIdx.x * blockDim.x + threadIdx.x;
    if (i >= total) return;
    const int n = i >> dlog;
    const int o = i & ((1 << dlog) - 1);
    const float nn = norm[n];
    agg[i] = xw[i] * nn * nn + b[o];
}

__global__ void k_edge_scatter(const int* __restrict__ ei, const float* __restrict__ xw,
                               const float* __restrict__ norm, float* __restrict__ agg,
                               int dlog, int E)
{
    const int t = blockIdx.x * blockDim.x + threadIdx.x;
    if (t >= (E << dlog)) return;
    const int e = t >> dlog;
    const int o = t & ((1 << dlog) - 1);
    const int s = ei[e];
    const int d = ei[E + e];
    atomicAdd(&agg[((size_t)d << dlog) + o],
              xw[((size_t)s << dlog) + o] * norm[s] * norm[d]);
}

// ============================================================================
// K: decoder — one LSTM cell per direction per row
// ============================================================================
__global__ void k_decoder(const int* __restrict__ x,
                          const float* __restrict__ hg, const float* __restrict__ cg,
                          const float* __restrict__ wihf, const float* __restrict__ whhf,
                          const float* __restrict__ bf,
                          const float* __restrict__ wihb, const float* __restrict__ whhb,
                          const float* __restrict__ bb,
                          float* __restrict__ dec_out)
{
    const int n = blockIdx.x * blockDim.x + threadIdx.x;
    if (n >= NROWS) return;
    const float start = (n == 0) ? -1.0f : (float)x[(n - 1) * SEQL + (SEQL - 1)];
#pragma unroll
    for (int d = 0; d < 2; ++d) {
        const float* wih = d ? wihb : wihf;
        const float* whh = d ? whhb : whhf;
        const float* bp  = d ? bb   : bf;
        float h0[HID], c0[HID];
#pragma unroll
        for (int j = 0; j < HID; ++j) {
            h0[j] = hg[(size_t)n * DD2 + d * HID + j];
            c0[j] = cg[(size_t)n * DD2 + d * HID + j];
        }
        float gates[G4];
        for (int g = 0; g < G4; ++g) {
            float acc = start * wih[g] + bp[g];
            for (int k = 0; k < HID; ++k) acc += h0[k] * whh[g * HID + k];
            gates[g] = acc;
        }
#pragma unroll
        for (int j = 0; j < HID; ++j) {
            const float iv = sigf(gates[j]);
            const float fv = sigf(gates[HID + j]);
            const float gv = tanhfast(gates[2 * HID + j]);
            const float ov = sigf(gates[3 * HID + j]);
            const float c1 = fv * c0[j] + iv * gv;
            dec_out[(size_t)n * DD2 + d * HID + j] = tanhfast(c1) * ov;
        }
    }
}

// ============================================================================
// K: final projection pred = dec_out @ out_w^T + out_b  (store-BW bound)
// f32 WMMA 16x16x4, K=26 padded to 28 (7 k-steps, tail lanes zeroed).
// Wave owns a 16x128 strip: A kept in regs, 8 N-subtiles.
// ============================================================================
__global__ void k_out_wmma(const float* __restrict__ A, const float* __restrict__ W,
                           const float* __restrict__ bias, float* __restrict__ out)
{
    const int gid  = blockIdx.x * blockDim.x + threadIdx.x;
    const int wave = gid >> 5;
    const int lane = threadIdx.x & 31;
    const int col  = lane & 15;
    const int koff = (lane & 16) >> 3;
    const int mt   = wave / 250;           // 256 M-tiles
    const int ng   = wave - mt * 250;      // 250 groups of 8 N-tiles
    const int m    = mt * 16 + col;

    v2f av[7];
    {
        const float* arow = A + (size_t)m * DD2;
#pragma unroll
        for (int s = 0; s < 7; ++s) {
            const int k = s * 4 + koff;
            const float a0 = arow[(k     < DD2) ? k     : 0];
            const float a1 = arow[(k + 1 < DD2) ? k + 1 : 0];
            av[s].x = (k     < DD2) ? a0 : 0.f;
            av[s].y = (k + 1 < DD2) ? a1 : 0.f;
        }
    }
    const int mrow = mt * 16 + ((lane & 16) >> 1);
#pragma unroll
    for (int sub = 0; sub < 8; ++sub) {
        const int ncol = (ng * 8 + sub) * 16 + col;
        const float* wrow = W + (size_t)ncol * DD2;
        v8f acc = {0.f, 0.f, 0.f, 0.f, 0.f, 0.f, 0.f, 0.f};
#pragma unroll
        for (int s = 0; s < 7; ++s) {
            const int k = s * 4 + koff;
            const float b0 = wrow[(k     < DD2) ? k     : 0];
            const float b1 = wrow[(k + 1 < DD2) ? k + 1 : 0];
            v2f bv; bv.x = (k < DD2) ? b0 : 0.f; bv.y = (k + 1 < DD2) ? b1 : 0.f;
            acc = __builtin_amdgcn_wmma_f32_16x16x4_f32(
                false, av[s], false, bv, (short)0, acc, false, false);
        }
        const float bn = bias[ncol];
#pragma unroll
        for (int v = 0; v < 8; ++v)
            out[(size_t)(mrow + v) * VOCAB + ncol] = acc[v] + bn;
    }
}

// ============================================================================
extern "C" void kernel_launch(void* const* d_in, const int* in_sizes, int n_in,
                              void* d_out, int out_size, void* d_ws, size_t ws_size,
                              hipStream_t stream)
{
    const int*   x     = (const int*)  d_in[0];
    const int*   ei    = (const int*)  d_in[1];
    const float* emb   = (const float*)d_in[2];
    const float* wihf  = (const float*)d_in[3];
    const float* whhf  = (const float*)d_in[4];
    const float* bf    = (const float*)d_in[5];
    const float* wihb  = (const float*)d_in[6];
    // d_in[7] = enc_w_hh_b: unused (backward LSTM only runs its first step, h0=0)
    const float* bb    = (const float*)d_in[8];
    const float* p1w   = (const float*)d_in[9];
    const float* p1b   = (const float*)d_in[10];
    const float* p2w   = (const float*)d_in[11];
    const float* p2b   = (const float*)d_in[12];
    const float* gh_w1 = (const float*)d_in[13];
    const float* gh_b1 = (const float*)d_in[14];
    const float* gh_w2 = (const float*)d_in[15];
    const float* gh_b2 = (const float*)d_in[16];
    const float* gh_fw = (const float*)d_in[17];
    const float* gh_fb = (const float*)d_in[18];
    const float* gc_w1 = (const float*)d_in[19];
    const float* gc_b1 = (const float*)d_in[20];
    const float* gc_w2 = (const float*)d_in[21];
    const float* gc_b2 = (const float*)d_in[22];
    const float* gc_fw = (const float*)d_in[23];
    const float* gc_fb = (const float*)d_in[24];
    const float* dwihf = (const float*)d_in[25];
    const float* dwhhf = (const float*)d_in[26];
    const float* dbf   = (const float*)d_in[27];
    const float* dwihb = (const float*)d_in[28];
    const float* dwhhb = (const float*)d_in[29];
    const float* dbb   = (const float*)d_in[30];
    const float* outw  = (const float*)d_in[31];
    const float* outb  = (const float*)d_in[32];

    char* ws = (char*)d_ws;
    _Float16* XG   = (_Float16*)(ws + OFF_XG);
    float*    LAST = (float*)(ws + OFF_LAST);
    float*    SH   = (float*)(ws + OFF_SH);
    float*    SC   = (float*)(ws + OFF_SC);
    float*    DEG  = (float*)(ws + OFF_DEG);
    float*    NORM = (float*)(ws + OFF_NORM);
    float*    XW   = (float*)(ws + OFF_XW);
    float*    AGG  = (float*)(ws + OFF_AGG);
    float*    HG   = (float*)(ws + OFF_HG);
    float*    CG   = (float*)(ws + OFF_CG);
    float*    DEC  = (float*)(ws + OFF_DEC);

    const int E = in_sizes[1] / 2;

    // encoder input gates (WMMA) + forward scan + backward step/projections
    k_xg_wmma<<<2048, 256, 0, stream>>>(x, emb, wihf, bf, XG);        // 16384 waves
    k_scan_fwd<<<NROWS / 256, 256, 0, stream>>>(XG, whhf, LAST);
    k_bwd_proj<<<NROWS / 256, 256, 0, stream>>>(x, emb, wihb, bb, p1w, p1b, p2w, p2b,
                                                LAST, SH, SC);

    // graph norm
    k_deg_init<<<NROWS / 256, 256, 0, stream>>>(DEG);
    k_deg_scatter<<<(E + 255) / 256, 256, 0, stream>>>(ei, DEG, E);
    k_norm<<<NROWS / 256, 256, 0, stream>>>(DEG, NORM);

    // GNN branch h (sh -> HG)
    k_rowmm<DD2, 16, false><<<NROWS / 256, 256, 0, stream>>>(SH, gh_w1, nullptr, XW);
    k_agg_init<<<(NROWS * 16 + 255) / 256, 256, 0, stream>>>(XW, NORM, gh_b1, AGG, 4, NROWS * 16);
    k_edge_scatter<<<((E << 4) + 255) / 256, 256, 0, stream>>>(ei, XW, NORM, AGG, 4, E);
    k_rowmm<16, 32, true><<<NROWS / 256, 256, 0, stream>>>(AGG, gh_w2, nullptr, XW);
    k_agg_init<<<(NROWS * 32 + 255) / 256, 256, 0, stream>>>(XW, NORM, gh_b2, AGG, 5, NROWS * 32);
    k_edge_scatter<<<((E << 5) + 255) / 256, 256, 0, stream>>>(ei, XW, NORM, AGG, 5, E);
    k_rowmm<32, DD2, true><<<NROWS / 256, 256, 0, stream>>>(AGG, gh_fw, gh_fb, HG);

    // GNN branch c (sc -> CG), reusing XW/AGG
    k_rowmm<DD2, 16, false><<<NROWS / 256, 256, 0, stream>>>(SC, gc_w1, nullptr, XW);
    k_agg_init<<<(NROWS * 16 + 255) / 256, 256, 0, stream>>>(XW, NORM, gc_b1, AGG, 4, NROWS * 16);
    k_edge_scatter<<<((E << 4) + 255) / 256, 256, 0, stream>>>(ei, XW, NORM, AGG, 4, E);
    k_rowmm<16, 32, true><<<NROWS / 256, 256, 0, stream>>>(AGG, gc_w2, nullptr, XW);
    k_agg_init<<<(NROWS * 32 + 255) / 256, 256, 0, stream>>>(XW, NORM, gc_b2, AGG, 5, NROWS * 32);
    k_edge_scatter<<<((E << 5) + 255) / 256, 256, 0, stream>>>(ei, XW, NORM, AGG, 5, E);
    k_rowmm<32, DD2, true><<<NROWS / 256, 256, 0, stream>>>(AGG, gc_fw, gc_fb, CG);

    // decoder + output projection (WMMA, store-bandwidth bound)
    k_decoder<<<NROWS / 256, 256, 0, stream>>>(x, HG, CG, dwihf, dwhhf, dbf,
                                               dwihb, dwhhb, dbb, DEC);
    k_out_wmma<<<8000, 256, 0, stream>>>(DEC, outw, outb, (float*)d_out);  // 64000 waves
    (void)n_in; (void)out_size; (void)ws_size;
}